// GHMC_69131793596448
// MI455X (gfx1250) — compile-verified
//
#include <hip/hip_runtime.h>

#define BINS 30
#define NBLK 1024
#define TPB 256
#define HSTRIDE 257              // 256 threads + 1 pad -> conflict-free banks

typedef float v2f __attribute__((ext_vector_type(2)));
typedef float v4f __attribute__((ext_vector_type(4)));
typedef int   v4i __attribute__((ext_vector_type(4)));
typedef float v8f __attribute__((ext_vector_type(8)));

// Pass 1: stream pred/target once, accumulate per-thread-private (bce_sum, count)
// histograms in LDS (zero-contention ds_add_f32), write one 64-wide partial row
// per block: cols [0..29]=bce sums, [30,31]=0, [32..61]=counts, [62,63]=0.
__global__ __launch_bounds__(TPB) void ghmc_pass(const float* __restrict__ pred,
                                                 const int* __restrict__ targ,
                                                 float* __restrict__ partials,
                                                 int nv4, int ntail) {
  __shared__ float hist[2 * BINS * HSTRIDE];   // ~60 KB of the 320 KB WGP LDS
  const int tid = threadIdx.x;
  for (int i = tid; i < 2 * BINS * HSTRIDE; i += TPB) hist[i] = 0.0f;
  __syncthreads();

  const v4f* p4 = (const v4f*)pred;
  const v4i* t4 = (const v4i*)targ;
  const int stride = gridDim.x * TPB;

  for (int i = blockIdx.x * TPB + tid; i < nv4; i += stride) {
    v4f p  = __builtin_nontemporal_load(p4 + i);   // read-once: NT, keep L2 clean
    v4i tt = __builtin_nontemporal_load(t4 + i);
#pragma unroll
    for (int k = 0; k < 4; ++k) {
      float x   = p[k];
      bool tone = (tt[k] != 0);
      float ax  = __builtin_fabsf(x);
      float e   = __expf(-ax);                     // v_exp_f32
      float r   = __builtin_amdgcn_rcpf(1.0f + e); // v_rcp_f32
      float a   = e * r;                           // e/(1+e)
      bool pos  = (x >= 0.0f);
      // g = |sigmoid(x)-t| collapses to a or 1-a by (sign(x)==t)
      float g   = (pos == tone) ? a : (1.0f - a);
      int idx   = (int)(g * (float)BINS);
      idx = idx > (BINS - 1) ? (BINS - 1) : idx;
      // stable bce = log1p(exp(-|x|)) + (|x| iff sign(x)!=t)
      float bce = __logf(1.0f + e) + ((pos != tone) ? ax : 0.0f);
      // privatized slots: single owner per address -> uncontended ds_add_f32
      atomicAdd(&hist[idx * HSTRIDE + tid], bce);
      atomicAdd(&hist[(idx + BINS) * HSTRIDE + tid], 1.0f);
    }
  }
  // scalar tail (block 0 only), if total elements not divisible by 4
  if (blockIdx.x == 0 && tid < ntail) {
    int e0 = nv4 * 4 + tid;
    float x = pred[e0];
    bool tone = (targ[e0] != 0);
    float ax = __builtin_fabsf(x);
    float e = __expf(-ax);
    float a = e * __builtin_amdgcn_rcpf(1.0f + e);
    bool pos = (x >= 0.0f);
    float g = (pos == tone) ? a : (1.0f - a);
    int idx = (int)(g * (float)BINS);
    idx = idx > (BINS - 1) ? (BINS - 1) : idx;
    float bce = __logf(1.0f + e) + ((pos != tone) ? ax : 0.0f);
    atomicAdd(&hist[idx * HSTRIDE + tid], bce);
    atomicAdd(&hist[(idx + BINS) * HSTRIDE + tid], 1.0f);
  }
  __syncthreads();

  // per-block reduce: 64 lanes, lane owns one output column; (s+i)%64 banks -> clean
  if (tid < 64) {
    int col, slot;
    if (tid < BINS)          { col = tid;     slot = tid; }   // bce sums  -> 0..29
    else if (tid < 2 * BINS) { col = tid + 2; slot = tid; }   // counts    -> 32..61
    else if (tid == 60)      { col = 30;      slot = -1; }    // zero pads
    else if (tid == 61)      { col = 31;      slot = -1; }
    else                     { col = tid;     slot = -1; }    // 62, 63
    float s = 0.0f;
    if (slot >= 0) {
      const float* h = &hist[slot * HSTRIDE];
      for (int i = 0; i < TPB; ++i) s += h[i];
    }
    partials[blockIdx.x * 64 + col] = s;
  }
}

// Pass 2: reduce nblk x 64 partials with V_WMMA_F32_16X16X4_F32 (A=ones so
// D[0,n] += sum_k B[k,n]); 4 waves own {sum,count} x {bins 0-15, bins 16-31}.
// Then thread 0 applies the GHM momentum/weight math on 30 bins.
__global__ __launch_bounds__(128) void ghmc_final(const float* __restrict__ partials,
                                                  const float* __restrict__ acc_sum,
                                                  float* __restrict__ out,
                                                  int nblk, float tot) {
  __shared__ float totals[64];
  const int lane = threadIdx.x & 31;
  const int w    = threadIdx.x >> 5;
  const int base = (w & 1) * 16 + (w >> 1) * 32;  // series base column
  const int col  = lane & 15;
  const int rofs = (lane >> 4) << 1;              // lanes 16-31 hold K=2,3 rows

  v2f aa; aa[0] = 1.0f; aa[1] = 1.0f;             // A = ones(16x4)
  v8f c = {};
  for (int j = 0; j < nblk; j += 4) {
    const float* row = partials + (j + rofs) * 64 + base + col;
    v2f bb;
    bb[0] = row[0];                               // K = rofs   row
    bb[1] = row[64];                              // K = rofs+1 row
    c = __builtin_amdgcn_wmma_f32_16x16x4_f32(false, aa, false, bb,
                                              (short)0, c, false, false);
  }
  if (lane < 16) totals[base + lane] = c[0];      // M=0 row lives in VGPR0, lanes 0-15
  __syncthreads();

  if (threadIdx.x == 0) {
    int nn = 0;
    for (int b = 0; b < BINS; ++b)
      if (totals[32 + b] > 0.0f) nn++;
    float nf = (float)(nn > 0 ? nn : 1);
    float loss = 0.0f;
    for (int b = 0; b < BINS; ++b) {
      float cnt = totals[32 + b];
      if (cnt > 0.0f) {
        float sacc = 0.75f * acc_sum[b] + 0.25f * cnt;   // MMT momentum blend
        // weights w_b = tot/sacc/nf; loss = sum(bce*w)/tot  ->  tot cancels
        loss += totals[b] / (sacc * nf);
      }
    }
    out[0] = loss;   // LOSS_WEIGHT = 1.0
    (void)tot;
  }
}

extern "C" void kernel_launch(void* const* d_in, const int* in_sizes, int n_in,
                              void* d_out, int out_size, void* d_ws, size_t ws_size,
                              hipStream_t stream) {
  const float* pred = (const float*)d_in[0];
  const int*   targ = (const int*)d_in[1];
  const float* acc  = (const float*)d_in[2];
  float* out        = (float*)d_out;
  float* partials   = (float*)d_ws;

  const int total = in_sizes[0];        // N*C = 32,000,000
  const int nv4   = total / 4;
  const int ntail = total - nv4 * 4;

  int nblk = NBLK;
  size_t need = (size_t)nblk * 64 * sizeof(float);
  if (ws_size < need) {                 // clamp (multiple of 4) if ws is small
    nblk = (int)(ws_size / (64 * sizeof(float))) & ~3;
    if (nblk < 4) nblk = 4;
  }

  ghmc_pass<<<nblk, TPB, 0, stream>>>(pred, targ, partials, nv4, ntail);
  ghmc_final<<<1, 128, 0, stream>>>(partials, acc, out, nblk, (float)total);
}